// OccupancyPooling_25348896981747
// MI455X (gfx1250) — compile-verified
//
#include <hip/hip_runtime.h>

#define GRID_G     6
#define HIDDEN     128
#define BLK        256
#define JT         2048
#define OCC_STRIDE 38   // >=36, even (8B-aligned v2f LDS loads), odd/64-ish for banks

typedef __attribute__((ext_vector_type(2))) float v2f;
typedef __attribute__((ext_vector_type(8))) float v8f;

__global__ __launch_bounds__(BLK)
void occ_embed_kernel(const float* __restrict__ obs,
                      const float* __restrict__ W,
                      const float* __restrict__ bias,
                      float* __restrict__ out,
                      int N)
{
    __shared__ float sObs[JT * 2];
    __shared__ float sOcc[BLK * OCC_STRIDE];

    const int tid = threadIdx.x;
    const int i   = blockIdx.x * BLK + tid;

    const float xi = obs[2 * i];
    const float yi = obs[2 * i + 1];

    // ---- Phase 1: per-thread private occupancy histogram in LDS ----
    float* myOcc = &sOcc[tid * OCC_STRIDE];
#pragma unroll
    for (int c = 0; c < OCC_STRIDE; ++c) myOcc[c] = 0.0f;

    for (int jt = 0; jt < N; jt += JT) {
        const int lim = (N - jt < JT) ? (N - jt) : JT;
        __syncthreads();
        for (int idx = tid; idx < lim; idx += BLK) {
            v2f o = *(const v2f*)(obs + 2 * (jt + idx));
            *(v2f*)(&sObs[2 * idx]) = o;
        }
        __syncthreads();
        for (int jj = 0; jj < lim; ++jj) {
            // rel = (obs_j - obs_i)/0.5 + 3 ; in-range iff both coords in [0,6)
            // NaN in either position fails all compares -> weight 0 (matches ref)
            float rx = (sObs[2 * jj]     - xi) * 2.0f + 3.0f;
            float ry = (sObs[2 * jj + 1] - yi) * 2.0f + 3.0f;
            bool inr = (rx >= 0.0f) & (rx < 6.0f) & (ry >= 0.0f) & (ry < 6.0f)
                       & ((jt + jj) != i);
            if (inr) {
                int cell = (int)rx * GRID_G + (int)ry;   // trunc==floor: rx,ry >= 0
                myOcc[cell] += 1.0f;                     // private region: no races
            }
        }
    }
    __syncthreads();

    // ---- Phase 2: out[256x128] = occ[256x36] @ W.T + b via V_WMMA_F32_16X16X4_F32 ----
    const int wave   = tid >> 5;          // wave32
    const int lane   = tid & 31;
    const int nloc   = lane & 15;         // N (and A's M) within tile
    const int khalf  = (lane >> 4) * 2;   // K pair selector
    const int rowb_g = blockIdx.x * BLK;

    for (int tile = wave; tile < 128; tile += 8) {   // 16 Mtiles x 8 Ntiles
        const int mtile = tile >> 3;
        const int ntile = tile & 7;
        const int arow  = mtile * 16 + nloc;         // A: M = lane%16
        const int hcol  = ntile * 16 + nloc;         // B: N = lane%16

        v8f acc = {};
#pragma unroll
        for (int step = 0; step < 9; ++step) {       // K = 36 = 9 x 4
            const int kb = step * 4 + khalf;         // even -> 8B aligned
            v2f a  = *(const v2f*)(&sOcc[arow * OCC_STRIDE + kb]);
            v2f bf = *(const v2f*)(W + hcol * 36 + kb);
            acc = __builtin_amdgcn_wmma_f32_16x16x4_f32(
                /*neg_a=*/false, a, /*neg_b=*/false, bf,
                /*c_mod=*/(short)0, acc, /*reuse_a=*/false, /*reuse_b=*/false);
        }

        const float bv = bias[hcol];
        const int mofs = (lane < 16) ? 0 : 8;        // D: lanes 16-31 hold M=8..15
#pragma unroll
        for (int r = 0; r < 8; ++r) {
            const int row = rowb_g + mtile * 16 + r + mofs;
            out[row * HIDDEN + ntile * 16 + nloc] = acc[r] + bv;
        }
    }
}

extern "C" void kernel_launch(void* const* d_in, const int* in_sizes, int n_in,
                              void* d_out, int out_size, void* d_ws, size_t ws_size,
                              hipStream_t stream) {
    const float* obs  = (const float*)d_in[0];   // [N,2]
    const float* W    = (const float*)d_in[1];   // [128,36]
    const float* bias = (const float*)d_in[2];   // [128]
    float* out = (float*)d_out;                  // [N,128]

    const int N = in_sizes[0] / 2;               // 8192
    const int blocks = (N + BLK - 1) / BLK;      // 32
    occ_embed_kernel<<<blocks, BLK, 0, stream>>>(obs, W, bias, out, N);
}